// RendererNaive_32134945309176
// MI455X (gfx1250) — compile-verified
//
#include <hip/hip_runtime.h>

// Gaussian-splat render: out[n,c] = sum_m exp(-0.5 * d^T Sigma_m^-1 d) * col[m,c]
// Rank-6 bilinear decomposition: maha = F(pix,6) @ G^T(6,gauss), run on
// V_WMMA_F32_16X16X4_F32 (K padded to 8 -> 2 WMMAs per 16x16 tile).

#define N_PIX   65536
#define N_GAUSS 2048
#define BLOCK   256          // 8 waves of 32
#define WAVES_PER_BLOCK 8
#define PIX_PER_WAVE 16
#define G_TILES (N_GAUSS / 16)

typedef __attribute__((ext_vector_type(2))) float v2f;
typedef __attribute__((ext_vector_type(8))) float v8f;

__global__ __launch_bounds__(BLOCK)
void render_wmma_kernel(const float* __restrict__ x,     // [N_PIX,2]
                        const float* __restrict__ mus,   // [N_GAUSS,2]
                        const float* __restrict__ covs,  // [N_GAUSS,2,2]
                        const float* __restrict__ cols,  // [N_GAUSS,3]
                        float* __restrict__ out)         // [N_PIX,3]
{
    // Per-Gaussian feature vectors g[0..7] (K padded to 8), 64 KB LDS.
    __shared__ float sG[N_GAUSS * 8];

    const int tid = threadIdx.x;

    // ---- Stage 1: precompute Gaussian features into LDS (once per block) ----
    for (int g = tid; g < N_GAUSS; g += BLOCK) {
        const float c00 = covs[4 * g + 0];
        const float c01 = covs[4 * g + 1];
        const float c11 = covs[4 * g + 3];
        const float inv = 1.0f / (c00 * c11 - c01 * c01);
        const float a  =  c11 * inv;            // Sigma^-1 = [[a,b],[b,c]]
        const float b  = -c01 * inv;
        const float c  =  c00 * inv;
        const float mx = mus[2 * g + 0];
        const float my = mus[2 * g + 1];
        const float b2 = 2.0f * b;
        sG[8 * g + 0] = a;
        sG[8 * g + 1] = b2;
        sG[8 * g + 2] = c;
        sG[8 * g + 3] = -2.0f * (a * mx + b * my);
        sG[8 * g + 4] = -(b2 * mx + 2.0f * c * my);
        sG[8 * g + 5] = a * mx * mx + b2 * mx * my + c * my * my;
        sG[8 * g + 6] = 0.0f;
        sG[8 * g + 7] = 0.0f;
    }
    __syncthreads();

    // ---- Stage 2: each wave renders 16 pixels against all 2048 Gaussians ----
    const int lane    = tid & 31;
    const int wave    = tid >> 5;
    const int pixBase = (blockIdx.x * WAVES_PER_BLOCK + wave) * PIX_PER_WAVE;
    const int row     = lane & 15;        // pixel row (A) / gaussian col (B)
    const bool hi     = lane >= 16;       // lane half selects K pair

    // Pixel features f = [px^2, px*py, py^2, px, py, 1, 0, 0]
    const float px = x[2 * (pixBase + row) + 0];
    const float py = x[2 * (pixBase + row) + 1];
    const float f0 = px * px, f1 = px * py, f2 = py * py;

    // A fragments (16x4 f32 layout: lanes 0-15 carry K0,K1; lanes 16-31 K2,K3)
    v2f a0, a1;
    a0.x = hi ? f2 : f0;       // K2 : K0
    a0.y = hi ? px : f1;       // K3 : K1
    a1.x = hi ? 0.0f : py;     // K6 : K4
    a1.y = hi ? 0.0f : 1.0f;   // K7 : K5

    float acc[8][3];
#pragma unroll
    for (int k = 0; k < 8; ++k) {
        acc[k][0] = 0.0f; acc[k][1] = 0.0f; acc[k][2] = 0.0f;
    }

    const float NEG_HALF_LOG2E = -0.7213475204444817f; // -0.5 * log2(e)

    for (int t = 0; t < G_TILES; ++t) {
        const int g = t * 16 + row;

        // B fragments from LDS (4x16 f32: lanes 0-15 K0,K1; lanes 16-31 K2,K3)
        const float* gp = sG + 8 * g + (hi ? 2 : 0);
        const float2 t0 = *reinterpret_cast<const float2*>(gp);      // ds_load_b64
        const float2 t1 = *reinterpret_cast<const float2*>(gp + 4);  // ds_load_b64
        v2f b0, b1;
        b0.x = t0.x; b0.y = t0.y;
        b1.x = t1.x; b1.y = t1.y;

        // maha tile (16 pix x 16 gauss) = A(16x8) * B(8x16) via two K=4 WMMAs
        v8f m = {0.f, 0.f, 0.f, 0.f, 0.f, 0.f, 0.f, 0.f};
        m = __builtin_amdgcn_wmma_f32_16x16x4_f32(false, a0, false, b0,
                                                  (short)0, m, false, false);
        m = __builtin_amdgcn_wmma_f32_16x16x4_f32(false, a1, false, b1,
                                                  (short)0, m, false, false);

        // This lane owns gaussian g for the whole tile -> load its RGB once.
        const float c0 = cols[3 * g + 0];
        const float c1 = cols[3 * g + 1];
        const float c2 = cols[3 * g + 2];

        // w = exp(-0.5*maha); accumulate RGB per pixel row (M = k + 8*hi).
#pragma unroll
        for (int k = 0; k < 8; ++k) {
            const float w = __builtin_amdgcn_exp2f(NEG_HALF_LOG2E * m[k]);
            acc[k][0] = __builtin_fmaf(w, c0, acc[k][0]);
            acc[k][1] = __builtin_fmaf(w, c1, acc[k][1]);
            acc[k][2] = __builtin_fmaf(w, c2, acc[k][2]);
        }
    }

    // ---- Stage 3: reduce over the 16 gaussian lanes in each half-wave ----
#pragma unroll
    for (int k = 0; k < 8; ++k) {
#pragma unroll
        for (int c = 0; c < 3; ++c) {
            float v = acc[k][c];
            v += __shfl_xor(v, 1);
            v += __shfl_xor(v, 2);
            v += __shfl_xor(v, 4);
            v += __shfl_xor(v, 8);   // xor masks < 16 stay inside each half
            acc[k][c] = v;
        }
    }

    // lanes 0 and 16 hold full sums for pixel rows 0-7 and 8-15 respectively
    if ((lane & 15) == 0) {
        const int pr = pixBase + (hi ? 8 : 0);
#pragma unroll
        for (int k = 0; k < 8; ++k) {
            out[3 * (pr + k) + 0] = acc[k][0];
            out[3 * (pr + k) + 1] = acc[k][1];
            out[3 * (pr + k) + 2] = acc[k][2];
        }
    }
}

extern "C" void kernel_launch(void* const* d_in, const int* in_sizes, int n_in,
                              void* d_out, int out_size, void* d_ws, size_t ws_size,
                              hipStream_t stream) {
    (void)in_sizes; (void)n_in; (void)out_size; (void)d_ws; (void)ws_size;
    const float* x    = (const float*)d_in[0];
    const float* mus  = (const float*)d_in[1];
    const float* covs = (const float*)d_in[2];
    const float* cols = (const float*)d_in[3];
    float* out = (float*)d_out;

    const int pixTiles = N_PIX / PIX_PER_WAVE;              // 4096 waves
    const int blocks   = pixTiles / WAVES_PER_BLOCK;        // 512 blocks
    render_wmma_kernel<<<blocks, BLOCK, 0, stream>>>(x, mus, covs, cols, out);
}